// SparseAttention_19413252178161
// MI455X (gfx1250) — compile-verified
//
#include <hip/hip_runtime.h>
#include <math.h>

// Problem constants (match reference)
#define NB     4       // batch
#define S_LEN  4096    // sequence length
#define D_DIM  512     // embed dim
#define BLK    128     // block size
#define NQB    32      // S_LEN / BLK

#define SCALE   0.04419417382415922f   // 1/sqrt(512)
#define MASKVAL (-1.0e9f)

typedef __attribute__((ext_vector_type(2))) float v2f;
typedef __attribute__((ext_vector_type(8))) float v8f;

// -------------------------------------------------------------------------
// Kernel 1: block-sparse scores + online softmax stats + in-place
//           normalization into the attn-weights region of d_out.
// grid = NB * NQB * 2 (each WG: 64 query rows), block = 128 (4 waves)
// Each wave owns 16 query rows; per key-block it computes a 16x128 score
// tile with V_WMMA_F32_16X16X4_F32, K staged in LDS.
// -------------------------------------------------------------------------
__global__ __launch_bounds__(128) void sparse_attn_scores(
    const float* __restrict__ Q, const float* __restrict__ K,
    float* __restrict__ attn) {
  const int wg   = blockIdx.x;
  const int half = wg & 1;
  const int qb   = (wg >> 1) & (NQB - 1);
  const int b    = wg >> 6;

  const int t    = threadIdx.x;
  const int lane = t & 31;
  const int wave = t >> 5;
  const int lrow = lane & 15;   // M / N index within tile
  const int grp  = lane >> 4;   // half-wave group (selects K pair / row+8)

  const int r0 = qb * BLK + half * 64 + wave * 16;  // wave's first query row

  const float* Qb = Q    + (size_t)b * S_LEN * D_DIM;
  const float* Kb = K    + (size_t)b * S_LEN * D_DIM;
  float*       Ab = attn + (size_t)b * S_LEN * S_LEN;

  __shared__ float ldsK[128 * 36];   // K chunk: 128 rows x 32 depth, stride 36
  __shared__ float m_sh[64];
  __shared__ float l_sh[64];

  float m_run[8], l_run[8];
#pragma unroll
  for (int i = 0; i < 8; ++i) { m_run[i] = -3.0e38f; l_run[i] = 0.0f; }

  const int kbmax = (qb + 1 < NQB) ? (qb + 1) : (NQB - 1);

  for (int kb = 0; kb <= kbmax; ++kb) {
    v8f acc[8] = {};
    const float* Krow = Kb + (size_t)(kb * BLK) * D_DIM;

    for (int dc = 0; dc < D_DIM; dc += 32) {
      __syncthreads();
      // Cooperative load: K[128][dc..dc+31] -> ldsK (row stride 36 dwords)
      {
        const int c4 = t & 7;    // which float4 in the 32-wide depth chunk
        const int rr = t >> 3;   // starting row (16 rows per pass)
#pragma unroll
        for (int r = rr; r < 128; r += 16) {
          float4 v = *(const float4*)(Krow + (size_t)r * D_DIM + dc + c4 * 4);
          *(float4*)(&ldsK[r * 36 + c4 * 4]) = v;
        }
      }
      // Prefetch next depth chunk of this K block (global_prefetch_b8)
      if (dc + 32 < D_DIM)
        __builtin_prefetch(Krow + (size_t)(t) * D_DIM + dc + 32, 0, 1);
      __syncthreads();

#pragma unroll
      for (int s8 = 0; s8 < 8; ++s8) {
        const int d = dc + s8 * 4;
        // A-frag (16x4 f32): lane = row M, K pair selected by group
        v2f a = *(const v2f*)(Qb + (size_t)(r0 + lrow) * D_DIM + d + 2 * grp);
#pragma unroll
        for (int n = 0; n < 8; ++n) {
          // B-frag (4x16 f32): lane = key row N, contiguous K pair in LDS
          v2f bf = *(const v2f*)(&ldsK[(n * 16 + lrow) * 36 + s8 * 4 + 2 * grp]);
          acc[n] = __builtin_amdgcn_wmma_f32_16x16x4_f32(
              false, a, false, bf, (short)0, acc[n], false, false);
        }
      }
    }

    // Epilogue: scale, mask, store raw scores, update online (m, l)
    const int colbase = kb * BLK;
#pragma unroll
    for (int i = 0; i < 8; ++i) {
      const int row = r0 + i + 8 * grp;
      float sv[8];
      float tmax = -3.0e38f;
#pragma unroll
      for (int n = 0; n < 8; ++n) {
        const int col = colbase + n * 16 + lrow;
        float s = acc[n][i] * SCALE;
        if (col > row + (BLK - 1)) s = MASKVAL;
        sv[n] = s;
        tmax  = fmaxf(tmax, s);
        Ab[(size_t)row * S_LEN + col] = s;   // raw score (normalized later)
      }
#pragma unroll
      for (int off = 1; off < 16; off <<= 1)
        tmax = fmaxf(tmax, __shfl_xor(tmax, off, 32));
      const float mnew = fmaxf(m_run[i], tmax);
      float ps = 0.0f;
#pragma unroll
      for (int n = 0; n < 8; ++n) ps += __expf(sv[n] - mnew);
#pragma unroll
      for (int off = 1; off < 16; off <<= 1)
        ps += __shfl_xor(ps, off, 32);
      l_run[i] = l_run[i] * __expf(m_run[i] - mnew) + ps;
      m_run[i] = mnew;
    }
  }

  // Publish per-row stats to LDS
  if ((lane & 15) == 0) {
#pragma unroll
    for (int i = 0; i < 8; ++i) {
      const int lr = wave * 16 + grp * 8 + i;
      m_sh[lr] = m_run[i];
      l_sh[lr] = l_run[i];
    }
  }
  __syncthreads();

  // Zero-fill fully-masked key blocks (final weights are exactly 0 there)
  const int zstart = (kbmax + 1) * BLK;
  if (zstart < S_LEN) {
    const int ncol4   = (S_LEN - zstart) >> 2;
    const int rowbase = qb * BLK + half * 64;
    const float4 z4   = make_float4(0.f, 0.f, 0.f, 0.f);
    for (int r = 0; r < 64; ++r) {
      float* dst = &Ab[(size_t)(rowbase + r) * S_LEN + zstart];
      for (int c = t; c < ncol4; c += 128) *(float4*)(dst + c * 4) = z4;
    }
  }

  // Normalize own rows in place: p = exp(s - m) / l  (same-wave data only)
  const int kend4 = zstart >> 2;
  for (int r = 0; r < 16; ++r) {
    const int   row = r0 + r;
    const float mm  = m_sh[wave * 16 + r];
    const float il  = 1.0f / l_sh[wave * 16 + r];
    float* base = &Ab[(size_t)row * S_LEN];
    for (int c = lane; c < kend4; c += 32) {
      float4 s4 = *(float4*)(base + c * 4);
      s4.x = __expf(s4.x - mm) * il;
      s4.y = __expf(s4.y - mm) * il;
      s4.z = __expf(s4.z - mm) * il;
      s4.w = __expf(s4.w - mm) * il;
      *(float4*)(base + c * 4) = s4;
    }
  }
}

// -------------------------------------------------------------------------
// Kernel 2: O = P * V with f32 WMMA. P and V chunks staged in LDS
// (V transposed so B-frags are aligned b64 LDS loads).
// grid = NB * NQB * 4 (each WG: 128 rows x 128 d-cols), block = 256 (8 waves)
// -------------------------------------------------------------------------
__global__ __launch_bounds__(256) void sparse_attn_pv(
    const float* __restrict__ attn, const float* __restrict__ V,
    float* __restrict__ Out) {
  const int wg     = blockIdx.x;
  const int dchunk = wg & 3;
  const int qb     = (wg >> 2) & (NQB - 1);
  const int b      = wg >> 7;

  const int t    = threadIdx.x;
  const int lane = t & 31;
  const int wave = t >> 5;
  const int lrow = lane & 15;
  const int grp  = lane >> 4;

  const int dbase = dchunk * 128;
  const int r0    = qb * BLK + wave * 16;

  const float* Ab = attn + (size_t)b * S_LEN * S_LEN;
  const float* Vb = V    + (size_t)b * S_LEN * D_DIM;
  float*       Ob = Out  + (size_t)b * S_LEN * D_DIM;

  __shared__ float ldsP[128 * 36];  // P chunk: 128 rows x 32 keys, stride 36
  __shared__ float ldsV[128 * 34];  // V chunk transposed: [col][k], stride 34

  const int kbmax = (qb + 1 < NQB) ? (qb + 1) : (NQB - 1);
  const int kend  = (kbmax + 1) * BLK;

  v8f acc[8] = {};

  for (int kc = 0; kc < kend; kc += 32) {
    __syncthreads();
    // Stage P: attn[qb*128 + r][kc .. kc+31]
    {
      const int rr = t >> 1;          // 0..127
      const int hh = (t & 1) * 16;    // k offset 0 or 16
      const float* src = Ab + (size_t)(qb * BLK + rr) * S_LEN + kc + hh;
      float*       dst = &ldsP[rr * 36 + hh];
#pragma unroll
      for (int c4 = 0; c4 < 4; ++c4)
        *(float4*)(dst + c4 * 4) = *(const float4*)(src + c4 * 4);
    }
    // Stage V transposed: V[kc+kr][dbase+c] -> ldsV[c*34 + kr]
    {
      const int c  = t & 127;
      const int k0 = t >> 7;          // 0 or 1
#pragma unroll
      for (int kr = k0; kr < 32; kr += 2)
        ldsV[c * 34 + kr] = Vb[(size_t)(kc + kr) * D_DIM + dbase + c];
    }
    __syncthreads();

#pragma unroll
    for (int s8 = 0; s8 < 8; ++s8) {
      v2f a = *(const v2f*)(&ldsP[(wave * 16 + lrow) * 36 + s8 * 4 + 2 * grp]);
#pragma unroll
      for (int n = 0; n < 8; ++n) {
        v2f bf = *(const v2f*)(&ldsV[(n * 16 + lrow) * 34 + s8 * 4 + 2 * grp]);
        acc[n] = __builtin_amdgcn_wmma_f32_16x16x4_f32(
            false, a, false, bf, (short)0, acc[n], false, false);
      }
    }
  }

  // Store output tile
#pragma unroll
  for (int i = 0; i < 8; ++i) {
    const int row = r0 + i + 8 * grp;
#pragma unroll
    for (int n = 0; n < 8; ++n) {
      const int col = dbase + n * 16 + lrow;
      Ob[(size_t)row * D_DIM + col] = acc[n][i];
    }
  }
}

// -------------------------------------------------------------------------
extern "C" void kernel_launch(void* const* d_in, const int* in_sizes, int n_in,
                              void* d_out, int out_size, void* d_ws,
                              size_t ws_size, hipStream_t stream) {
  const float* Q = (const float*)d_in[0];
  const float* K = (const float*)d_in[1];
  const float* V = (const float*)d_in[2];

  float* out  = (float*)d_out;                          // [4,4096,512]
  float* attn = out + (size_t)NB * S_LEN * D_DIM;       // [4,4096,4096]

  sparse_attn_scores<<<NB * NQB * 2, 128, 0, stream>>>(Q, K, attn);
  sparse_attn_pv<<<NB * NQB * 4, 256, 0, stream>>>(attn, V, out);
}